// RelativeSelfMultiheadAttn_1786706395353
// MI455X (gfx1250) — compile-verified
//
#include <hip/hip_runtime.h>
#include <hip/hip_bf16.h>

// RelativeSelfMultiheadAttn for MI455X (gfx1250, wave32).
// All GEMMs use V_WMMA_F32_16X16X4_F32 (full f32 precision; the workload is
// memory-bound on the 256MB attn output, so fp32 matrix math is free).
// Big GEMMs use 2x2 (or 1x4) register tile blocking -> 1:1 load:wmma ratio.
//
// Shapes: T=1024 B=4 D=1024 H=16 Dh=64 R=2047.

typedef float v2f __attribute__((ext_vector_type(2)));
typedef float v8f __attribute__((ext_vector_type(8)));

#define WMMA_F32(a, b, c) \
  __builtin_amdgcn_wmma_f32_16x16x4_f32(false, (a), false, (b), (short)0, (c), false, false)

#define T_LEN 1024
#define B_SZ 4
#define D_MODEL 1024
#define N_HEAD 16
#define DH 64
#define R_LEN 2047   // 2T-1

// ---------------------------------------------------------------------------
// 1) LayerNorm: q[T,B,D] -> X[T,B,D].  One block per row of 1024.
// ---------------------------------------------------------------------------
__global__ void ln_kernel(const float* __restrict__ q, const float* __restrict__ w,
                          const float* __restrict__ b, float* __restrict__ x) {
  const int row = blockIdx.x;                       // t*B + b, 4096 rows
  const float* qr = q + (size_t)row * D_MODEL;
  float* xr = x + (size_t)row * D_MODEL;
  __shared__ float red[256];
  float v[4];
  float s = 0.f;
#pragma unroll
  for (int i = 0; i < 4; ++i) { v[i] = qr[threadIdx.x + 256 * i]; s += v[i]; }
  red[threadIdx.x] = s; __syncthreads();
  for (int off = 128; off > 0; off >>= 1) {
    if (threadIdx.x < off) red[threadIdx.x] += red[threadIdx.x + off];
    __syncthreads();
  }
  const float mu = red[0] * (1.0f / D_MODEL);
  __syncthreads();
  s = 0.f;
#pragma unroll
  for (int i = 0; i < 4; ++i) { float d = v[i] - mu; s += d * d; }
  red[threadIdx.x] = s; __syncthreads();
  for (int off = 128; off > 0; off >>= 1) {
    if (threadIdx.x < off) red[threadIdx.x] += red[threadIdx.x + off];
    __syncthreads();
  }
  const float rstd = rsqrtf(red[0] * (1.0f / D_MODEL) + 1e-5f);
#pragma unroll
  for (int i = 0; i < 4; ++i) {
    int c = threadIdx.x + 256 * i;
    xr[c] = (v[i] - mu) * rstd * w[c] + b[c];
  }
}

// ---------------------------------------------------------------------------
// Epilogue helper for the fused QKV GEMM: scatter one 16x16 C-tile into
// qw (=q+r_w_bias), qr (=q+r_r_bias), K or V in [B,H,T,Dh] layout.
// ---------------------------------------------------------------------------
__device__ __forceinline__ void qkv_store_tile(
    const v8f& acc, int m16, int cbase, int lane,
    const float* __restrict__ bias, const float* __restrict__ rwb,
    const float* __restrict__ rrb,
    float* __restrict__ qw, float* __restrict__ qr_,
    float* __restrict__ kh, float* __restrict__ vh) {
  const int n = lane & 15, mbase = (lane >> 4) * 8;
  const int c = cbase + n;                          // column in [0,3072)
  const int h = c / 192, rem = c % 192, sel = rem / 64, dh = rem % 64;
  const float bv = bias[c];
  const float rw = rwb[h * DH + dh], rr = rrb[h * DH + dh];
#pragma unroll
  for (int i = 0; i < 8; ++i) {
    const int row = m16 + mbase + i;                // = t*B + b
    const int t = row >> 2, bb = row & 3;
    const float val = acc[i] + bv;
    const size_t off = ((size_t)((bb * N_HEAD + h) * T_LEN + t)) * DH + dh;
    if (sel == 0) { qw[off] = val + rw; qr_[off] = val + rr; }
    else if (sel == 1) kh[off] = val;
    else vh[off] = val;
  }
}

// ---------------------------------------------------------------------------
// 2) Fused QKV GEMM: X[4096,1024] @ Win[3072,1024]^T + bias.
//    2x2 register blocking: each wave computes a 32x32 output macro-tile,
//    4 fragment loads feed 4 WMMAs per k-step (1:1 load:wmma).
// ---------------------------------------------------------------------------
__global__ void qkv_gemm(const float* __restrict__ X, const float* __restrict__ W,
                         const float* __restrict__ bias,
                         const float* __restrict__ rwb, const float* __restrict__ rrb,
                         float* __restrict__ qw, float* __restrict__ qr_,
                         float* __restrict__ kh, float* __restrict__ vh) {
  const int lane = threadIdx.x & 31, wave = threadIdx.x >> 5;
  const int mt = blockIdx.y * 8 + wave;             // 0..127 (32-row macro tiles)
  const int nt = blockIdx.x;                        // 0..95  (32-col macro tiles)
  const int m0 = mt * 32, n0 = nt * 32;
  const int ml = lane & 15, k2 = (lane >> 4) * 2;
  const float* Ap0 = X + (size_t)(m0 + ml) * D_MODEL + k2;
  const float* Ap1 = Ap0 + 16 * D_MODEL;
  const float* Bp0 = W + (size_t)(n0 + ml) * D_MODEL + k2;
  const float* Bp1 = Bp0 + 16 * D_MODEL;
  v8f acc00 = {}, acc01 = {}, acc10 = {}, acc11 = {};
  for (int k = 0; k < D_MODEL; k += 4) {
    v2f a0 = *(const v2f*)(Ap0 + k);
    v2f a1 = *(const v2f*)(Ap1 + k);
    v2f b0 = *(const v2f*)(Bp0 + k);
    v2f b1 = *(const v2f*)(Bp1 + k);
    acc00 = WMMA_F32(a0, b0, acc00);
    acc01 = WMMA_F32(a0, b1, acc01);
    acc10 = WMMA_F32(a1, b0, acc10);
    acc11 = WMMA_F32(a1, b1, acc11);
  }
  qkv_store_tile(acc00, m0,      n0,      lane, bias, rwb, rrb, qw, qr_, kh, vh);
  qkv_store_tile(acc01, m0,      n0 + 16, lane, bias, rwb, rrb, qw, qr_, kh, vh);
  qkv_store_tile(acc10, m0 + 16, n0,      lane, bias, rwb, rrb, qw, qr_, kh, vh);
  qkv_store_tile(acc11, m0 + 16, n0 + 16, lane, bias, rwb, rrb, qw, qr_, kh, vh);
}

// ---------------------------------------------------------------------------
// 3) Position GEMM: pos[2047,1024] @ Wpos[1024,1024]^T + bias -> rpos[2047,1024]
//    (row-major == [R,H,Dh]).  Row 2047 of the padded tile is guarded.
// ---------------------------------------------------------------------------
__global__ void pos_gemm(const float* __restrict__ P, const float* __restrict__ W,
                         const float* __restrict__ bias, float* __restrict__ rpos) {
  const int lane = threadIdx.x & 31, wave = threadIdx.x >> 5;
  const int mt = blockIdx.y * 8 + wave;             // 0..127 (rows 0..2047)
  const int nt = blockIdx.x;                        // 0..63
  const int m0 = mt * 16, n0 = nt * 16;
  const int ml = lane & 15, k2 = (lane >> 4) * 2;
  const int arow = m0 + ml;
  const bool avalid = arow < R_LEN;
  const float* Ap = P + (size_t)(avalid ? arow : 0) * D_MODEL + k2;
  const float* Bp = W + (size_t)(n0 + ml) * D_MODEL + k2;
  const v2f zero = {0.f, 0.f};
  v8f acc = {};
  for (int k = 0; k < D_MODEL; k += 4) {
    v2f a = zero;
    if (avalid) a = *(const v2f*)(Ap + k);
    v2f b = *(const v2f*)(Bp + k);
    acc = WMMA_F32(a, b, acc);
  }
  const int n = lane & 15, mbase = (lane >> 4) * 8;
  const float bv = bias[n0 + n];
#pragma unroll
  for (int i = 0; i < 8; ++i) {
    const int row = m0 + mbase + i;
    if (row < R_LEN) rpos[(size_t)row * D_MODEL + n0 + n] = acc[i] + bv;
  }
}

// ---------------------------------------------------------------------------
// 4) Scores: S[b,h,t,s] = 0.125*( qw[t]·K[s] + qr[t]·rpos[(T-1)-t+s] ) + mask.
//    Per wave: one 16x16 tile. AC via 16 WMMAs; BD via a 16x32 tile (two
//    accumulators over rpos rows jbase..jbase+31), rel-shifted through LDS.
// ---------------------------------------------------------------------------
__global__ void score_kernel(const float* __restrict__ qw, const float* __restrict__ qr,
                             const float* __restrict__ kh, const float* __restrict__ rpos,
                             const unsigned char* __restrict__ mask,
                             float* __restrict__ attn) {
  __shared__ float dtile[8][16][32];
  const int lane = threadIdx.x & 31, wave = threadIdx.x >> 5;
  const int bh = blockIdx.z;                        // b*16 + h
  const int b = bh >> 4, h = bh & 15;
  const int tt = blockIdx.y * 8 + wave;             // 0..63
  const int ss = blockIdx.x;                        // 0..63
  const int t0 = tt * 16, s0 = ss * 16;
  const int ml = lane & 15, k2 = (lane >> 4) * 2;
  const float* qwp = qw + ((size_t)bh * T_LEN + t0 + ml) * DH + k2;
  const float* qrp = qr + ((size_t)bh * T_LEN + t0 + ml) * DH + k2;
  const float* kp  = kh + ((size_t)bh * T_LEN + s0 + ml) * DH + k2;
  const int jbase = (T_LEN - 16) - t0 + s0;         // 0..2016
  const int j0 = jbase + ml;                        // always < 2047
  const int j1 = jbase + 16 + ml;                   // may hit 2047 -> guard
  const bool v1ok = j1 < R_LEN;
  const float* rp0 = rpos + (size_t)j0 * D_MODEL + h * DH + k2;
  const float* rp1 = rpos + (size_t)(v1ok ? j1 : 0) * D_MODEL + h * DH + k2;
  const v2f zero = {0.f, 0.f};
  v8f ac = {}, d0 = {}, d1 = {};
  for (int k = 0; k < DH; k += 4) {
    v2f a  = *(const v2f*)(qwp + k);
    v2f bk = *(const v2f*)(kp + k);
    ac = WMMA_F32(a, bk, ac);
    v2f a2 = *(const v2f*)(qrp + k);
    v2f b0 = *(const v2f*)(rp0 + k);
    d0 = WMMA_F32(a2, b0, d0);
    v2f b1 = zero;
    if (v1ok) b1 = *(const v2f*)(rp1 + k);
    d1 = WMMA_F32(a2, b1, d1);
  }
  const int n = lane & 15, mbase = (lane >> 4) * 8;
#pragma unroll
  for (int i = 0; i < 8; ++i) {
    dtile[wave][mbase + i][n]      = d0[i];
    dtile[wave][mbase + i][16 + n] = d1[i];
  }
  // per-wave LDS region; DS ops are in-order within a wave -> no barrier.
  const float inv_sqrt = 0.125f;                    // 1/sqrt(Dh)
  const bool masked = mask[b * T_LEN + s0 + n] != 0;
  float* out = attn + ((size_t)bh * T_LEN + t0 + mbase) * T_LEN + s0 + n;
#pragma unroll
  for (int i = 0; i < 8; ++i) {
    const int m = mbase + i;
    float sc = (ac[i] + dtile[wave][m][15 - m + n]) * inv_sqrt;
    if (masked) sc = -1e9f;
    out[(size_t)i * T_LEN] = sc;
  }
}

// ---------------------------------------------------------------------------
// 5) Row softmax in place over attn[B*H*T, T].
// ---------------------------------------------------------------------------
__global__ void softmax_kernel(float* __restrict__ attn) {
  float* p = attn + (size_t)blockIdx.x * T_LEN;
  __shared__ float red[256];
  float v[4];
  float mx = -3.4e38f;
#pragma unroll
  for (int i = 0; i < 4; ++i) { v[i] = p[threadIdx.x + 256 * i]; mx = fmaxf(mx, v[i]); }
  red[threadIdx.x] = mx; __syncthreads();
  for (int off = 128; off > 0; off >>= 1) {
    if (threadIdx.x < off) red[threadIdx.x] = fmaxf(red[threadIdx.x], red[threadIdx.x + off]);
    __syncthreads();
  }
  mx = red[0]; __syncthreads();
  float s = 0.f;
#pragma unroll
  for (int i = 0; i < 4; ++i) { v[i] = expf(v[i] - mx); s += v[i]; }
  red[threadIdx.x] = s; __syncthreads();
  for (int off = 128; off > 0; off >>= 1) {
    if (threadIdx.x < off) red[threadIdx.x] += red[threadIdx.x + off];
    __syncthreads();
  }
  const float inv = 1.0f / red[0];
#pragma unroll
  for (int i = 0; i < 4; ++i) p[threadIdx.x + 256 * i] = v[i] * inv;
}

// ---------------------------------------------------------------------------
// 6) ctx GEMM per (b,h): attn[T,T] @ V[T,Dh] -> ctx2 in [T,B,D] layout.
//    1x4 N-blocking: each wave computes 16 x 64 (full Dh); the attn-row
//    fragment (the 256MB HBM stream) is loaded once and reused 4x.
// ---------------------------------------------------------------------------
__global__ void ctx_gemm(const float* __restrict__ attn, const float* __restrict__ vh,
                         float* __restrict__ ctx2) {
  const int lane = threadIdx.x & 31, wave = threadIdx.x >> 5;
  const int bh = blockIdx.z, b = bh >> 4, h = bh & 15;
  const int mt = blockIdx.x * 8 + wave;             // 0..63
  const int m0 = mt * 16;
  const int ml = lane & 15, k2 = (lane >> 4) * 2;
  const float* Ap = attn + ((size_t)bh * T_LEN + m0 + ml) * T_LEN + k2;
  const float* Vp = vh + (size_t)bh * T_LEN * DH;
  const int nb = lane & 15;
  v8f acc[4] = {{}, {}, {}, {}};
  for (int k = 0; k < T_LEN; k += 4) {
    v2f a = *(const v2f*)(Ap + k);
    const int kr = k + k2;
    const float* vr0 = Vp + (size_t)kr * DH + nb;
    const float* vr1 = vr0 + DH;
#pragma unroll
    for (int j = 0; j < 4; ++j) {
      v2f bv;
      bv.x = vr0[j * 16];
      bv.y = vr1[j * 16];
      acc[j] = WMMA_F32(a, bv, acc[j]);
    }
  }
  const int n = lane & 15, mbase = (lane >> 4) * 8;
#pragma unroll
  for (int j = 0; j < 4; ++j) {
#pragma unroll
    for (int i = 0; i < 8; ++i) {
      const int t = m0 + mbase + i;
      ctx2[((size_t)t * B_SZ + b) * D_MODEL + h * DH + j * 16 + n] = acc[j][i];
    }
  }
}

// ---------------------------------------------------------------------------
// 7) Output GEMM: ctx2[4096,1024] @ Wout[1024,1024]^T + bias + residual(q).
//    2x2 register blocking like qkv_gemm.
// ---------------------------------------------------------------------------
__global__ void out_gemm(const float* __restrict__ C2, const float* __restrict__ W,
                         const float* __restrict__ bias, const float* __restrict__ resid,
                         float* __restrict__ out) {
  const int lane = threadIdx.x & 31, wave = threadIdx.x >> 5;
  const int mt = blockIdx.y * 8 + wave;             // 0..127
  const int nt = blockIdx.x;                        // 0..31
  const int m0 = mt * 32, n0 = nt * 32;
  const int ml = lane & 15, k2 = (lane >> 4) * 2;
  const float* Ap0 = C2 + (size_t)(m0 + ml) * D_MODEL + k2;
  const float* Ap1 = Ap0 + 16 * D_MODEL;
  const float* Bp0 = W + (size_t)(n0 + ml) * D_MODEL + k2;
  const float* Bp1 = Bp0 + 16 * D_MODEL;
  v8f acc00 = {}, acc01 = {}, acc10 = {}, acc11 = {};
  for (int k = 0; k < D_MODEL; k += 4) {
    v2f a0 = *(const v2f*)(Ap0 + k);
    v2f a1 = *(const v2f*)(Ap1 + k);
    v2f b0 = *(const v2f*)(Bp0 + k);
    v2f b1 = *(const v2f*)(Bp1 + k);
    acc00 = WMMA_F32(a0, b0, acc00);
    acc01 = WMMA_F32(a0, b1, acc01);
    acc10 = WMMA_F32(a1, b0, acc10);
    acc11 = WMMA_F32(a1, b1, acc11);
  }
  const int n = lane & 15, mbase = (lane >> 4) * 8;
#pragma unroll
  for (int tj = 0; tj < 2; ++tj) {
    const int cc = n0 + tj * 16 + n;
    const float bv = bias[cc];
    const v8f* accs[2] = {tj == 0 ? &acc00 : &acc01, tj == 0 ? &acc10 : &acc11};
#pragma unroll
    for (int ti = 0; ti < 2; ++ti) {
      const v8f av = *accs[ti];
#pragma unroll
      for (int i = 0; i < 8; ++i) {
        const size_t idx = (size_t)(m0 + ti * 16 + mbase + i) * D_MODEL + cc;
        out[idx] = av[i] + bv + resid[idx];
      }
    }
  }
}

// ---------------------------------------------------------------------------
extern "C" void kernel_launch(void* const* d_in, const int* in_sizes, int n_in,
                              void* d_out, int out_size, void* d_ws, size_t ws_size,
                              hipStream_t stream) {
  (void)in_sizes; (void)n_in; (void)out_size; (void)ws_size;
  const float* q      = (const float*)d_in[0];   // [T,B,D]
  const float* pos    = (const float*)d_in[1];   // [2047,1,D] == [2047,D]
  const unsigned char* mask = (const unsigned char*)d_in[2]; // [B,T] bool
  const float* in_w   = (const float*)d_in[3];   // [3D,D]
  const float* in_b   = (const float*)d_in[4];
  const float* pos_w  = (const float*)d_in[5];   // [D,D]
  const float* pos_b  = (const float*)d_in[6];
  const float* out_w  = (const float*)d_in[7];   // [D,D]
  const float* out_b  = (const float*)d_in[8];
  const float* rwb    = (const float*)d_in[9];   // [H,Dh]
  const float* rrb    = (const float*)d_in[10];
  const float* lnw    = (const float*)d_in[11];
  const float* lnb    = (const float*)d_in[12];

  float* out_p  = (float*)d_out;                          // [T,B,D] = 4M floats
  float* attn_p = (float*)d_out + (size_t)T_LEN * B_SZ * D_MODEL; // [B,H,T,T]

  // Workspace layout (floats). Total 22M floats = 88 MB.
  constexpr size_t MFL = 1048576;
  float* ws   = (float*)d_ws;
  float* X    = ws;               // 4M; dead after qkv_gemm -> reused as ctx2
  float* QW   = ws + 4 * MFL;     // 4M  q + r_w_bias, [B,H,T,Dh]
  float* QR   = ws + 8 * MFL;     // 4M  q + r_r_bias
  float* KH   = ws + 12 * MFL;    // 4M
  float* VH   = ws + 16 * MFL;    // 4M
  float* RPOS = ws + 20 * MFL;    // 2M  (2047 used rows, row 2047 guarded)
  float* CTX2 = X;

  ln_kernel<<<T_LEN * B_SZ, 256, 0, stream>>>(q, lnw, lnb, X);
  qkv_gemm<<<dim3(96, 16, 1), 256, 0, stream>>>(X, in_w, in_b, rwb, rrb, QW, QR, KH, VH);
  pos_gemm<<<dim3(64, 16, 1), 256, 0, stream>>>(pos, pos_w, pos_b, RPOS);
  score_kernel<<<dim3(64, 8, B_SZ * N_HEAD), 256, 0, stream>>>(QW, QR, KH, RPOS, mask, attn_p);
  softmax_kernel<<<B_SZ * N_HEAD * T_LEN, 256, 0, stream>>>(attn_p);
  ctx_gemm<<<dim3(8, 1, B_SZ * N_HEAD), 256, 0, stream>>>(attn_p, VH, CTX2);
  out_gemm<<<dim3(32, 16, 1), 256, 0, stream>>>(CTX2, out_w, out_b, q, out_p);
}